// SpatialGather_Module_4664334483623
// MI455X (gfx1250) — compile-verified
//
#include <hip/hip_runtime.h>
#include <hip/hip_bf16.h>
#include <math.h>

typedef __attribute__((ext_vector_type(2))) float v2f;
typedef __attribute__((ext_vector_type(8))) float v8f;

#define SCALE_F 1.0f

constexpr int Bn = 8;        // batch
constexpr int Kk = 19;       // prob rows
constexpr int Cc = 512;      // channels
constexpr int Nn = 32768;    // spatial (128*256)
constexpr int KPAD = 32;     // K padded to 2 WMMA tiles
constexpr int NCHUNK = 1024; // n range per workgroup
constexpr int NTILE = 64;    // staged n tile
constexpr int CCHUNK = 128;  // c range per workgroup
constexpr int LDSS = NTILE + 4; // padded LDS row stride (68 floats = 272B, 16B aligned)
constexpr int TILES = NCHUNK / NTILE; // 16

// -------- CDNA5 async global->LDS path (guarded; sync fallback keeps compile safe) ------
#if defined(__HIP_DEVICE_COMPILE__) && __has_builtin(__builtin_amdgcn_global_load_async_to_lds_b128)
#define USE_ASYNC 1
typedef int v4i_raw __attribute__((vector_size(16)));
typedef __attribute__((address_space(1))) v4i_raw gas_v4i;  // global int4
typedef __attribute__((address_space(3))) v4i_raw las_v4i;  // LDS int4
#else
#define USE_ASYNC 0
#endif

template <int N>
__device__ __forceinline__ void wait_async() {
#if USE_ASYNC
#if __has_builtin(__builtin_amdgcn_s_wait_asynccnt)
    __builtin_amdgcn_s_wait_asynccnt(N);
#else
    asm volatile("s_wait_asynccnt %0" ::"i"(N) : "memory");
#endif
#endif
}

// ---------------- Phase 1: per-(b,k) softmax stats (max, 1/sum) ----------------
__global__ __launch_bounds__(256) void softmax_stats_kernel(const float* __restrict__ probs,
                                                            float* __restrict__ stats) {
    const int row = blockIdx.x;              // b*Kk + k
    const float* p = probs + (size_t)row * Nn;
    __shared__ float red[256];
    const int tid = threadIdx.x;

    float m = -INFINITY;
    for (int i = tid; i < Nn; i += 256) m = fmaxf(m, SCALE_F * p[i]);
    red[tid] = m; __syncthreads();
    for (int s = 128; s > 0; s >>= 1) {
        if (tid < s) red[tid] = fmaxf(red[tid], red[tid + s]);
        __syncthreads();
    }
    m = red[0]; __syncthreads();

    float sum = 0.f;
    for (int i = tid; i < Nn; i += 256) sum += expf(SCALE_F * p[i] - m);
    red[tid] = sum; __syncthreads();
    for (int s = 128; s > 0; s >>= 1) {
        if (tid < s) red[tid] += red[tid + s];
        __syncthreads();
    }
    if (tid == 0) {
        stats[row * 2]     = m;
        stats[row * 2 + 1] = 1.0f / red[0];
    }
}

// ---------------- Phase 2 helpers ----------------
__device__ __forceinline__ void issue_feats_tile(const float* __restrict__ gbase,
                                                 float* lbuf, int tid) {
    const int col   = (tid & 15) * 4;
    const int rbase = tid >> 4;
#pragma unroll
    for (int j = 0; j < 8; ++j) {
        const int r = rbase + j * 16;
        const float* gp = gbase + (size_t)r * Nn + col;
        float* lp = &lbuf[r * LDSS + col];
#if USE_ASYNC
        // 8 async b128 ops per wave per tile, DMA straight into LDS (ASYNCcnt)
        __builtin_amdgcn_global_load_async_to_lds_b128((gas_v4i*)gp, (las_v4i*)lp, 0, 0);
#else
        *(float4*)lp = *(const float4*)gp;
#endif
    }
}

__device__ __forceinline__ void stage_attn_tile(const float* __restrict__ pbase,
                                                float* abuf,
                                                const float* s_m, const float* s_is,
                                                int tid) {
#pragma unroll
    for (int j = 0; j < 2; ++j) {
        const int idx = tid + j * 256;
        const int k   = idx >> 4;
        const int col = (idx & 15) * 4;
        float4 v = make_float4(0.f, 0.f, 0.f, 0.f);
        if (k < Kk) {
            const float4 pv = *(const float4*)(pbase + (size_t)k * Nn + col);
            const float m = s_m[k], is = s_is[k];
            v.x = expf(SCALE_F * pv.x - m) * is;
            v.y = expf(SCALE_F * pv.y - m) * is;
            v.z = expf(SCALE_F * pv.z - m) * is;
            v.w = expf(SCALE_F * pv.w - m) * is;
        }
        *(float4*)(&abuf[k * LDSS + col]) = v;
    }
}

__device__ __forceinline__ void compute_tile(const float* fbuf, const float* abuf,
                                             int wave, int lane, v8f& acc0, v8f& acc1) {
    const int lsel = (lane & 16) ? 2 : 0;  // f32 A/B layout: lanes 16-31 hold K=2,3
    const float* frow = fbuf + (wave * 16 + (lane & 15)) * LDSS + lsel;
    const float* ar0  = abuf + (lane & 15) * LDSS + lsel;
    const float* ar1  = ar0 + 16 * LDSS;
#pragma unroll
    for (int s = 0; s < NTILE / 4; ++s) {
        const v2f av = *(const v2f*)(frow + s * 4);
        const v2f b0 = *(const v2f*)(ar0 + s * 4);
        const v2f b1 = *(const v2f*)(ar1 + s * 4);
        acc0 = __builtin_amdgcn_wmma_f32_16x16x4_f32(
            false, av, false, b0, (short)0, acc0, false, false);
        acc1 = __builtin_amdgcn_wmma_f32_16x16x4_f32(
            false, av, false, b1, (short)0, acc1, false, false);
    }
}

// ---------------- Phase 2: D[c,k] += feats-tile x attn-tile via WMMA f32 ----------------
__global__ __launch_bounds__(256) void gather_wmma_kernel(const float* __restrict__ feats,
                                                          const float* __restrict__ probs,
                                                          const float* __restrict__ stats,
                                                          float* __restrict__ out) {
    __shared__ float lds_f[2][CCHUNK * LDSS];  // double-buffered feats tiles
    __shared__ float lds_a[2][KPAD * LDSS];    // double-buffered attn tiles
    __shared__ float s_m[KPAD], s_is[KPAD];

    const int tid = threadIdx.x;
    const int b  = blockIdx.z;
    const int c0 = blockIdx.y * CCHUNK;
    const int n0 = blockIdx.x * NCHUNK;

    if (tid < Kk) {
        s_m[tid]  = stats[(b * Kk + tid) * 2];
        s_is[tid] = stats[(b * Kk + tid) * 2 + 1];
    }
    __syncthreads();  // stats visible before attn staging

    const float* fbase = feats + (size_t)(b * Cc + c0) * Nn + n0;
    const float* pbase = probs + (size_t)(b * Kk) * Nn + n0;

    v8f acc0 = {};  // k tile 0 (k = 0..15)
    v8f acc1 = {};  // k tile 1 (k = 16..31, only 16..18 valid)
    const int wave = tid >> 5;
    const int lane = tid & 31;

    // prologue: kick off tile 0
    issue_feats_tile(fbase, lds_f[0], tid);
    stage_attn_tile(pbase, lds_a[0], s_m, s_is, tid);

    for (int t = 0; t < TILES; ++t) {
        const int cur = t & 1;
        if (t + 1 < TILES) {
            // overlap: DMA tile t+1 while tile t computes
            issue_feats_tile(fbase + (t + 1) * NTILE, lds_f[cur ^ 1], tid);
            stage_attn_tile(pbase + (t + 1) * NTILE, lds_a[cur ^ 1], s_m, s_is, tid);
            wait_async<8>();   // tile t's 8 async ops done; tile t+1's may remain in flight
        } else {
            wait_async<0>();
        }
        __syncthreads();       // + dscnt drain for attn stores, all waves aligned
        compute_tile(lds_f[cur], lds_a[cur], wave, lane, acc0, acc1);
        __syncthreads();       // protect buffers before next overwrite
    }

    // ---- accumulate D[c,k] into out[b,c,k] (C/D layout: VGPR g -> M=g(+8), N=lane&15) ----
    const int mhalf = (lane >> 4) * 8;
    const int kcol  = lane & 15;
#pragma unroll
    for (int g = 0; g < 8; ++g) {
        const int c = c0 + wave * 16 + mhalf + g;
        atomicAdd(out + ((size_t)b * Cc + c) * Kk + kcol, acc0[g]);
        if (kcol < Kk - 16)
            atomicAdd(out + ((size_t)b * Cc + c) * Kk + 16 + kcol, acc1[g]);
    }
}

extern "C" void kernel_launch(void* const* d_in, const int* in_sizes, int n_in,
                              void* d_out, int out_size, void* d_ws, size_t ws_size,
                              hipStream_t stream) {
    const float* feats = (const float*)d_in[0];
    const float* probs = (const float*)d_in[1];
    float* out   = (float*)d_out;
    float* stats = (float*)d_ws;     // needs Bn*Kk*2 floats = 1216 B

    (void)hipMemsetAsync(d_out, 0, (size_t)out_size * sizeof(float), stream);

    softmax_stats_kernel<<<Bn * Kk, 256, 0, stream>>>(probs, stats);

    dim3 grid(Nn / NCHUNK, Cc / CCHUNK, Bn);   // 32 x 4 x 8 = 1024 WGs
    gather_wmma_kernel<<<grid, 256, 0, stream>>>(feats, probs, stats, out);
}